// Model_Geom_Graph_C_S_34153579938673
// MI455X (gfx1250) — compile-verified
//
#include <hip/hip_runtime.h>
#include <hip/hip_bf16.h>
#include <math.h>

#define NHEAD 16
#define SIZE_T 500
#define NEG_INF_KEY 0x007FFFFFu   // fkey(-inf)

typedef __attribute__((ext_vector_type(2))) float v2f;
typedef __attribute__((ext_vector_type(8))) float v8f;

__device__ __forceinline__ float lrelu_(float v, float s) { return v >= 0.f ? v : s * v; }
// order-preserving float<->uint key (works for negative values, so atomicMax(uint) == float max)
__device__ __forceinline__ unsigned fkey(float f) {
  unsigned b = __float_as_uint(f);
  return (b & 0x80000000u) ? ~b : (b | 0x80000000u);
}
__device__ __forceinline__ float fdec(unsigned k) {
  unsigned b = (k & 0x80000000u) ? (k & 0x7FFFFFFFu) : ~k;
  return __uint_as_float(b);
}

// ---------------- edge-attention helpers ----------------
// Q[h,k] = sum_c We[h*5+c,k] * a_e[h,c]  -> per-edge term is just ea @ Q^T
__global__ void k_prep_q(const float* __restrict__ We, const float* __restrict__ ae,
                         float* __restrict__ Q) {
  int t = threadIdx.x;
  if (t >= 32) return;
  int h = t >> 1, k = t & 1;
  float s = 0.f;
#pragma unroll
  for (int c = 0; c < 5; ++c) s += We[(h * 5 + c) * 2 + k] * ae[h * 5 + c];
  Q[h * 2 + k] = s;
}

__device__ __forceinline__ bool edge_setup(int e, int E, const int* __restrict__ ei,
                                           const float* __restrict__ ea, int use_mask,
                                           int& src, int& dst, float& e0, float& e1) {
  if (e < E) {
    src = ei[e]; dst = ei[E + e];
    e0 = ea[2 * e]; e1 = ea[2 * e + 1];
    if (use_mask && !(e0 > 1.0f)) return false;
    return true;
  }
  src = dst = e - E; e0 = e1 = 0.f;  // self loop, zero edge_attr, never masked
  return true;
}

__device__ __forceinline__ void edge_al(int src, int dst, float e0, float e1,
                                        const float* __restrict__ als,
                                        const float* __restrict__ ald,
                                        const float* __restrict__ Q, float al[NHEAD]) {
  const float4* ps = (const float4*)(als + (size_t)src * NHEAD);
  const float4* pd = (const float4*)(ald + (size_t)dst * NHEAD);
#pragma unroll
  for (int q = 0; q < 4; ++q) {
    float4 vs = ps[q], vd = pd[q];
    int h = q * 4;
    al[h + 0] = lrelu_(vs.x + vd.x + e0 * Q[2 * (h + 0)] + e1 * Q[2 * (h + 0) + 1], 0.2f);
    al[h + 1] = lrelu_(vs.y + vd.y + e0 * Q[2 * (h + 1)] + e1 * Q[2 * (h + 1) + 1], 0.2f);
    al[h + 2] = lrelu_(vs.z + vd.z + e0 * Q[2 * (h + 2)] + e1 * Q[2 * (h + 2) + 1], 0.2f);
    al[h + 3] = lrelu_(vs.w + vd.w + e0 * Q[2 * (h + 3)] + e1 * Q[2 * (h + 3) + 1], 0.2f);
  }
}

// ---------------- GAT layer 1: tiny K=5 transform + alpha dot-products ----------------
__global__ void k_gat1_node(const float* __restrict__ x, const float* __restrict__ W,
                            const float* __restrict__ a_s, const float* __restrict__ a_d,
                            float* __restrict__ xp, float* __restrict__ als,
                            float* __restrict__ ald, int* __restrict__ timing, int N) {
  __shared__ float sW[400], ss[80], sd[80];
  for (int i = threadIdx.x; i < 400; i += blockDim.x) sW[i] = W[i];
  for (int i = threadIdx.x; i < 80; i += blockDim.x) { ss[i] = a_s[i]; sd[i] = a_d[i]; }
  __syncthreads();
  int n = blockIdx.x * blockDim.x + threadIdx.x;
  if (n >= N) return;
  float xv[5];
#pragma unroll
  for (int c = 0; c < 5; ++c) xv[c] = x[(size_t)n * 5 + c];
  timing[n] = (int)xv[2];
#pragma unroll
  for (int h = 0; h < NHEAD; ++h) {
    float accs = 0.f, accd = 0.f;
#pragma unroll
    for (int c = 0; c < 5; ++c) {
      int o = h * 5 + c;
      float v = 0.f;
#pragma unroll
      for (int k = 0; k < 5; ++k) v += sW[o * 5 + k] * xv[k];
      xp[(size_t)n * 80 + o] = v;
      accs += v * ss[o];
      accd += v * sd[o];
    }
    als[(size_t)n * NHEAD + h] = accs;
    ald[(size_t)n * NHEAD + h] = accd;
  }
}

// alpha dot-products for layer 2 (xp already computed by WMMA GEMM)
__global__ void k_node_alpha(const float* __restrict__ xp, const float* __restrict__ a_s,
                             const float* __restrict__ a_d, float* __restrict__ als,
                             float* __restrict__ ald, int N) {
  __shared__ float ss[80], sd[80];
  for (int i = threadIdx.x; i < 80; i += blockDim.x) { ss[i] = a_s[i]; sd[i] = a_d[i]; }
  __syncthreads();
  int n = blockIdx.x * blockDim.x + threadIdx.x;
  if (n >= N) return;
  const float* p = xp + (size_t)n * 80;
#pragma unroll
  for (int h = 0; h < NHEAD; ++h) {
    float accs = 0.f, accd = 0.f;
#pragma unroll
    for (int c = 0; c < 5; ++c) {
      float v = p[h * 5 + c];
      accs += v * ss[h * 5 + c];
      accd += v * sd[h * 5 + c];
    }
    als[(size_t)n * NHEAD + h] = accs;
    ald[(size_t)n * NHEAD + h] = accd;
  }
}

// ---------------- WMMA f32 GEMM: D[N,80] = A[N,80] @ W[80,80]^T ----------------
// One wave per 16x16 output tile; 20 steps of V_WMMA_F32_16X16X4_F32 (full fp32 precision).
__global__ void __launch_bounds__(32) k_gemm_wmma80(const float* __restrict__ Am,
                                                    const float* __restrict__ Wm,
                                                    float* __restrict__ D, int N) {
  int tile = blockIdx.x;
  int rt = tile / 5;       // row tile (M)
  int ot = tile % 5;       // output-channel tile (N dim of GEMM)
  int lane = threadIdx.x;
  int half = lane >> 4;    // 0: K pair {k,k+1}; 1: {k+2,k+3} (ISA 16x4 f32 A layout)
  int l15 = lane & 15;
  int row = rt * 16 + l15;
  int rowc = row < N ? row : N - 1;           // keep EXEC all-ones (WMMA requirement)
  int ocol = ot * 16 + l15;
  v8f acc = {};
#pragma unroll
  for (int k = 0; k < 80; k += 4) {
    float2 fa = *(const float2*)(Am + (size_t)rowc * 80 + k + 2 * half);
    float2 fb = *(const float2*)(Wm + (size_t)ocol * 80 + k + 2 * half);  // B[kk,n]=W[n,kk]
    v2f a; a[0] = fa.x; a[1] = fa.y;
    v2f b; b[0] = fb.x; b[1] = fb.y;
    acc = __builtin_amdgcn_wmma_f32_16x16x4_f32(false, a, false, b, (short)0, acc,
                                                false, false);
  }
  // D tile: lanes 0-15 -> rows M=0..7 in vgpr r; lanes 16-31 -> M=8..15
  int crow0 = rt * 16 + 8 * half;
#pragma unroll
  for (int r = 0; r < 8; ++r) {
    int rr = crow0 + r;
    if (rr < N) D[(size_t)rr * 80 + ot * 16 + l15] = acc[r];
  }
}

// ---------------- segment softmax (3 edge passes) ----------------
__global__ void k_seg_init(unsigned* __restrict__ mkey, float* __restrict__ ssum, int n) {
  int i = blockIdx.x * blockDim.x + threadIdx.x;
  if (i < n) { mkey[i] = NEG_INF_KEY; ssum[i] = 0.f; }
}

__global__ void k_edge_max(const int* __restrict__ ei, const float* __restrict__ ea,
                           const float* __restrict__ Q, const float* __restrict__ als,
                           const float* __restrict__ ald, unsigned* __restrict__ mkey,
                           int E, int N, int use_mask) {
  int e = blockIdx.x * blockDim.x + threadIdx.x;
  if (e >= E + N) return;
  int src, dst; float e0, e1;
  if (!edge_setup(e, E, ei, ea, use_mask, src, dst, e0, e1)) return;
  float al[NHEAD];
  edge_al(src, dst, e0, e1, als, ald, Q, al);
  unsigned* mp = mkey + (size_t)dst * NHEAD;
#pragma unroll
  for (int h = 0; h < NHEAD; ++h) atomicMax(mp + h, fkey(al[h]));
}

__global__ void k_edge_sum(const int* __restrict__ ei, const float* __restrict__ ea,
                           const float* __restrict__ Q, const float* __restrict__ als,
                           const float* __restrict__ ald, const unsigned* __restrict__ mkey,
                           float* __restrict__ ssum, int E, int N, int use_mask) {
  int e = blockIdx.x * blockDim.x + threadIdx.x;
  if (e >= E + N) return;
  int src, dst; float e0, e1;
  if (!edge_setup(e, E, ei, ea, use_mask, src, dst, e0, e1)) return;
  float al[NHEAD];
  edge_al(src, dst, e0, e1, als, ald, Q, al);
  const unsigned* mp = mkey + (size_t)dst * NHEAD;
  float* sp = ssum + (size_t)dst * NHEAD;
#pragma unroll
  for (int h = 0; h < NHEAD; ++h) atomicAdd(sp + h, __expf(al[h] - fdec(mp[h])));
}

__global__ void k_bias_init(float* __restrict__ out, const float* __restrict__ b, int N) {
  int i = blockIdx.x * blockDim.x + threadIdx.x;
  if (i < N * 80) out[i] = b[i % 80];
}

__global__ void k_edge_agg(const int* __restrict__ ei, const float* __restrict__ ea,
                           const float* __restrict__ Q, const float* __restrict__ als,
                           const float* __restrict__ ald, const unsigned* __restrict__ mkey,
                           const float* __restrict__ ssum, const float* __restrict__ xp,
                           float* __restrict__ out, int E, int N, int use_mask) {
  int e = blockIdx.x * blockDim.x + threadIdx.x;
  if (e >= E + N) return;
  int src, dst; float e0, e1;
  if (!edge_setup(e, E, ei, ea, use_mask, src, dst, e0, e1)) return;
  float al[NHEAD];
  edge_al(src, dst, e0, e1, als, ald, Q, al);
  const unsigned* mp = mkey + (size_t)dst * NHEAD;
  const float* sp = ssum + (size_t)dst * NHEAD;
  const float* px = xp + (size_t)src * 80;
  float* po = out + (size_t)dst * 80;
#pragma unroll
  for (int h = 0; h < NHEAD; ++h) {
    float coef = __expf(al[h] - fdec(mp[h])) / sp[h];
#pragma unroll
    for (int c = 0; c < 5; ++c) atomicAdd(po + h * 5 + c, coef * px[h * 5 + c]);
  }
}

// ---------------- scatter-max pooling (timing bins) ----------------
__global__ void k_pool_init(unsigned* __restrict__ pkey, int n) {
  int i = blockIdx.x * blockDim.x + threadIdx.x;
  if (i < n) pkey[i] = NEG_INF_KEY;
}
__global__ void k_pool_scatter(const float* __restrict__ h2, const int* __restrict__ timing,
                               unsigned* __restrict__ pkey, int N, int nseg) {
  int idx = blockIdx.x * blockDim.x + threadIdx.x;
  if (idx >= N * 80) return;
  int n = idx / 80, k = idx % 80;
  int t = timing[n];
  t = t < 0 ? 0 : (t >= nseg ? nseg - 1 : t);
  atomicMax(pkey + (size_t)t * 80 + k, fkey(h2[idx]));
}
__global__ void k_pool_decode(const unsigned* __restrict__ pkey, float* __restrict__ z, int n) {
  int i = blockIdx.x * blockDim.x + threadIdx.x;
  if (i >= n) return;
  float v = fdec(pkey[i]);
  z[i] = isfinite(v) ? v : 0.f;   // empty segments -> 0 (ref semantics)
}

// ---------------- Conv1d(k=10, stride=2, pad=1) + LeakyReLU(0.01) ----------------
__global__ void k_conv(const float* __restrict__ in, const float* __restrict__ w,
                       const float* __restrict__ bias, float* __restrict__ out,
                       int B, int Cin, int Lin, int Cout, int Lout) {
  int idx = blockIdx.x * blockDim.x + threadIdx.x;
  if (idx >= B * Cout * Lout) return;
  int t = idx % Lout, rem = idx / Lout;
  int oc = rem % Cout, b = rem / Cout;
  float acc = bias[oc];
  int t0 = t * 2 - 1;
  const float* wrow = w + (size_t)oc * Cin * 10;
  const float* inb = in + (size_t)b * Cin * Lin;
  for (int ci = 0; ci < Cin; ++ci) {
    const float* ir = inb + (size_t)ci * Lin;
    const float* wr = wrow + ci * 10;
#pragma unroll
    for (int j = 0; j < 10; ++j) {
      int ti = t0 + j;
      if (ti >= 0 && ti < Lin) acc += ir[ti] * wr[j];
    }
  }
  out[idx] = lrelu_(acc, 0.01f);
}

__global__ void k_maxpool10(const float* __restrict__ in, float* __restrict__ out,
                            int B, int C, int L, int W) {
  int idx = blockIdx.x * blockDim.x + threadIdx.x;
  if (idx >= B * C * W) return;
  int w = idx % W, c = (idx / W) % C, b = idx / (W * C);
  const float* p = in + ((size_t)b * C + c) * L + w * 10;
  float m = p[0];
#pragma unroll
  for (int j = 1; j < 10; ++j) m = fmaxf(m, p[j]);
  out[((size_t)b * C + c) * W + w] = m;  // == flatten(B, C*W)
}

// ---------------- MLP head: 80 -> 128 -> 128 -> 1 (lrelu 0.01; optional tanh) ----------------
__global__ void __launch_bounds__(128) k_mlp(const float* __restrict__ z,
                                             const float* __restrict__ w1, const float* __restrict__ b1,
                                             const float* __restrict__ w2, const float* __restrict__ b2,
                                             const float* __restrict__ w3, const float* __restrict__ b3,
                                             float* __restrict__ out, int which, int do_tanh) {
  __shared__ float sz[80], h1s[128], h2s[128], red[128];
  int b = blockIdx.x, j = threadIdx.x;
  for (int i = j; i < 80; i += 128) sz[i] = z[(size_t)b * 80 + i];
  __syncthreads();
  float a = b1[j];
  for (int k = 0; k < 80; ++k) a += w1[(size_t)j * 80 + k] * sz[k];
  h1s[j] = lrelu_(a, 0.01f);
  __syncthreads();
  float a2 = b2[j];
  for (int k = 0; k < 128; ++k) a2 += w2[(size_t)j * 128 + k] * h1s[k];
  h2s[j] = lrelu_(a2, 0.01f);
  __syncthreads();
  red[j] = w3[j] * h2s[j];
  __syncthreads();
  for (int s = 64; s > 0; s >>= 1) {
    if (j < s) red[j] += red[j + s];
    __syncthreads();
  }
  if (j == 0) {
    float o = red[0] + b3[0];
    if (do_tanh) o = tanhf(o);
    out[b * 3 + which] = o;
  }
}

extern "C" void kernel_launch(void* const* d_in, const int* in_sizes, int n_in,
                              void* d_out, int out_size, void* d_ws, size_t ws_size,
                              hipStream_t stream) {
  (void)n_in; (void)ws_size;
  const float* x     = (const float*)d_in[0];
  const float* eattr = (const float*)d_in[1];
  const float* g1W  = (const float*)d_in[2];
  const float* g1We = (const float*)d_in[3];
  const float* g1as = (const float*)d_in[4];
  const float* g1ad = (const float*)d_in[5];
  const float* g1ae = (const float*)d_in[6];
  const float* g1b  = (const float*)d_in[7];
  const float* g2W  = (const float*)d_in[8];
  const float* g2We = (const float*)d_in[9];
  const float* g2as = (const float*)d_in[10];
  const float* g2ad = (const float*)d_in[11];
  const float* g2ae = (const float*)d_in[12];
  const float* g2b  = (const float*)d_in[13];
  const float* c1w = (const float*)d_in[14]; const float* c1b = (const float*)d_in[15];
  const float* c2w = (const float*)d_in[16]; const float* c2b = (const float*)d_in[17];
  const float* c3w = (const float*)d_in[18]; const float* c3b = (const float*)d_in[19];
  const float* mw[3][3], *mb[3][3];
  for (int m = 0; m < 3; ++m)
    for (int l = 0; l < 3; ++l) {
      mw[m][l] = (const float*)d_in[20 + m * 6 + l * 2];
      mb[m][l] = (const float*)d_in[20 + m * 6 + l * 2 + 1];
    }
  const int* eindex = (const int*)d_in[38];
  float* out = (float*)d_out;

  int N = in_sizes[0] / 5;
  int E = in_sizes[1] / 2;
  int B = out_size / 3;        // num_graphs
  int EP = E + N;

  // ---- carve workspace (~145 MB for N=100k, E=1.2M, B=64) ----
  char* wsb = (char*)d_ws;
  size_t off = 0;
  auto carve = [&](size_t bytes) -> void* {
    void* p = wsb + off;
    off += (bytes + 255) & ~(size_t)255;
    return p;
  };
  float*    xp     = (float*)carve((size_t)N * 80 * 4);
  float*    als    = (float*)carve((size_t)N * NHEAD * 4);
  float*    ald    = (float*)carve((size_t)N * NHEAD * 4);
  unsigned* mkey   = (unsigned*)carve((size_t)N * NHEAD * 4);
  float*    ssum   = (float*)carve((size_t)N * NHEAD * 4);
  float*    h1     = (float*)carve((size_t)N * 80 * 4);
  float*    h2     = (float*)carve((size_t)N * 80 * 4);
  int*      timing = (int*)carve((size_t)N * 4);
  float*    Q1     = (float*)carve(256);
  float*    Q2     = (float*)carve(256);
  int nseg = B * SIZE_T;
  unsigned* pkey = (unsigned*)carve((size_t)nseg * 80 * 4);
  float*    z    = (float*)carve((size_t)nseg * 80 * 4);   // == [B,80,500] flat
  int L1 = (SIZE_T + 2 - 10) / 2 + 1;   // 247
  int L2 = (L1 + 2 - 10) / 2 + 1;       // 120
  int L3 = (L2 + 2 - 10) / 2 + 1;       // 57
  int NW = (L3 - 10) / 10 + 1;          // 5
  float* z1 = (float*)carve((size_t)B * 16 * L1 * 4);
  float* z2 = (float*)carve((size_t)B * 16 * L2 * 4);
  float* z3 = (float*)carve((size_t)B * 16 * L3 * 4);
  float* zp = (float*)carve((size_t)B * 80 * 4);

  dim3 blk(256);
  auto G = [](long long n) { return dim3((unsigned)((n + 255) / 256)); };

  // precompute collapsed edge-attention matrices
  k_prep_q<<<1, 32, 0, stream>>>(g1We, g1ae, Q1);
  k_prep_q<<<1, 32, 0, stream>>>(g2We, g2ae, Q2);

  // ---- GAT layer 1 ----
  k_gat1_node<<<G(N), blk, 0, stream>>>(x, g1W, g1as, g1ad, xp, als, ald, timing, N);
  k_seg_init<<<G((long long)N * NHEAD), blk, 0, stream>>>(mkey, ssum, N * NHEAD);
  k_edge_max<<<G(EP), blk, 0, stream>>>(eindex, eattr, Q1, als, ald, mkey, E, N, 0);
  k_edge_sum<<<G(EP), blk, 0, stream>>>(eindex, eattr, Q1, als, ald, mkey, ssum, E, N, 0);
  k_bias_init<<<G((long long)N * 80), blk, 0, stream>>>(h1, g1b, N);
  k_edge_agg<<<G(EP), blk, 0, stream>>>(eindex, eattr, Q1, als, ald, mkey, ssum, xp, h1, E, N, 0);

  // ---- GAT layer 2: WMMA f32 GEMM for the node transform ----
  int rtiles = (N + 15) / 16;
  k_gemm_wmma80<<<dim3(rtiles * 5), dim3(32), 0, stream>>>(h1, g2W, xp, N);
  k_node_alpha<<<G(N), blk, 0, stream>>>(xp, g2as, g2ad, als, ald, N);
  k_seg_init<<<G((long long)N * NHEAD), blk, 0, stream>>>(mkey, ssum, N * NHEAD);
  k_edge_max<<<G(EP), blk, 0, stream>>>(eindex, eattr, Q2, als, ald, mkey, E, N, 1);
  k_edge_sum<<<G(EP), blk, 0, stream>>>(eindex, eattr, Q2, als, ald, mkey, ssum, E, N, 1);
  k_bias_init<<<G((long long)N * 80), blk, 0, stream>>>(h2, g2b, N);
  k_edge_agg<<<G(EP), blk, 0, stream>>>(eindex, eattr, Q2, als, ald, mkey, ssum, xp, h2, E, N, 1);

  // ---- max_pool_x over timing bins ----
  k_pool_init<<<G((long long)nseg * 80), blk, 0, stream>>>(pkey, nseg * 80);
  k_pool_scatter<<<G((long long)N * 80), blk, 0, stream>>>(h2, timing, pkey, N, nseg);
  k_pool_decode<<<G((long long)nseg * 80), blk, 0, stream>>>(pkey, z, nseg * 80);

  // ---- conv stack + maxpool ----
  k_conv<<<G((long long)B * 16 * L1), blk, 0, stream>>>(z, c1w, c1b, z1, B, 80, SIZE_T, 16, L1);
  k_conv<<<G((long long)B * 16 * L2), blk, 0, stream>>>(z1, c2w, c2b, z2, B, 16, L1, 16, L2);
  k_conv<<<G((long long)B * 16 * L3), blk, 0, stream>>>(z2, c3w, c3b, z3, B, 16, L2, 16, L3);
  k_maxpool10<<<G((long long)B * 16 * NW), blk, 0, stream>>>(z3, zp, B, 16, L3, NW);

  // ---- three MLP heads -> out [B,3] ----
  k_mlp<<<dim3(B), dim3(128), 0, stream>>>(zp, mw[0][0], mb[0][0], mw[0][1], mb[0][1],
                                           mw[0][2], mb[0][2], out, 0, 1);
  k_mlp<<<dim3(B), dim3(128), 0, stream>>>(zp, mw[1][0], mb[1][0], mw[1][1], mb[1][1],
                                           mw[1][2], mb[1][2], out, 1, 0);
  k_mlp<<<dim3(B), dim3(128), 0, stream>>>(zp, mw[2][0], mb[2][0], mw[2][1], mb[2][1],
                                           mw[2][2], mb[2][2], out, 2, 0);
}